// TopKGate_47648367182395
// MI455X (gfx1250) — compile-verified
//
#include <hip/hip_runtime.h>
#include <hip/hip_bf16.h>

typedef float v2f __attribute__((ext_vector_type(2)));
typedef float v4f __attribute__((ext_vector_type(4)));
typedef float v8f __attribute__((ext_vector_type(8)));

#define TOKENS   8192
#define DIM      4096
#define NEXP     64
#define BM       64      // tokens per block
#define KC       64      // K-chunk staged in LDS
#define PSTR     80      // pair-row stride (float2 units): 2*80 % 64 == 32 -> disjoint bank halves
#define LSTRIDE  72      // padded LDS stride for logits

__global__ __launch_bounds__(128) void
TopKGate_router_kernel(const float* __restrict__ x,
                       const float* __restrict__ w,
                       const float* __restrict__ b,
                       float* __restrict__ gates,
                       int* __restrict__ idxout)
{
    // B chunk in WMMA-fragment order: shB[kpair][expert] = (w[k], w[k+1])
    // -> each lane's B operand is ONE contiguous ds_load_b64, no shuffle movs.
    __shared__ v2f   shB[(KC / 2) * PSTR];
    __shared__ float shL[BM * LSTRIDE];     // logits tile: [token][expert]

    const int tid  = threadIdx.x;
    const int lane = tid & 31;
    const int wv   = tid >> 5;              // wave id 0..3
    const int tokBase = blockIdx.x * BM;

    // A-fragment addressing (V_WMMA_F32_16X16X4_F32):
    // lanes 0-15: M = lane, K = {0,1}; lanes 16-31: M = lane-16, K = {2,3}
    const int klane = lane >> 4;            // 0 or 1
    const int arow  = tokBase + wv * 16 + (lane & 15);
    const float* xrow = x + (size_t)arow * DIM + klane * 2;

    const int bcol = lane & 15;             // column within a 16-expert tile

    v8f acc[4] = {};                        // 4 tiles of 16x16 f32 accum

    // w -> LDS staging map: 2 threads per expert row, 32 K each
    const int we = tid >> 1;                // expert 0..63
    const int wk = (tid & 1) * 32;          // K offset within chunk
    const float* wp = w + (size_t)we * DIM + wk;

    for (int k0 = 0; k0 < DIM; k0 += KC) {
        // Stage w[*, k0:k0+KC] into LDS in fragment order (coalesced float4 reads)
        #pragma unroll
        for (int i = 0; i < 32; i += 4) {
            v4f v = *(const v4f*)(wp + k0 + i);
            v2f lo; lo.x = v.x; lo.y = v.y;
            v2f hi; hi.x = v.z; hi.y = v.w;
            const int r = (wk + i) >> 1;    // pair-row
            shB[(r + 0) * PSTR + we] = lo;
            shB[(r + 1) * PSTR + we] = hi;
        }
        __syncthreads();

        // Prefetch next x chunk while this one computes (global_prefetch_b8)
        if (k0 + KC < DIM)
            __builtin_prefetch(xrow + k0 + KC, 0, 1);

        #pragma unroll
        for (int kk = 0; kk < KC; kk += 4) {
            // x streamed once, no reuse -> non-temporal load keeps L2 for w
            v2f a = __builtin_nontemporal_load((const v2f*)(xrow + k0 + kk));
            const v2f* brow = &shB[((kk >> 1) + klane) * PSTR];
            #pragma unroll
            for (int t = 0; t < 4; ++t) {
                v2f bb = brow[t * 16 + bcol];     // single ds_load_b64
                acc[t] = __builtin_amdgcn_wmma_f32_16x16x4_f32(
                    false, a, false, bb, (short)0, acc[t], false, false);
            }
        }
        __syncthreads();
    }

    // Spill logits to LDS per C/D layout: VGPR j -> row j (lanes 0-15) / j+8 (16-31)
    {
        const int rbase = wv * 16 + klane * 8;
        #pragma unroll
        for (int t = 0; t < 4; ++t) {
            #pragma unroll
            for (int j = 0; j < 8; ++j)
                shL[(rbase + j) * LSTRIDE + t * 16 + (lane & 15)] = acc[t][j];
        }
    }
    __syncthreads();

    // Per-token bias add + top-2 (strict '>' => first-occurrence ties, like lax.top_k)
    if (tid < BM) {
        const float* lrow = &shL[tid * LSTRIDE];
        float v0 = -3.402823466e38f, v1 = -3.402823466e38f;
        int   i0 = 0, i1 = 0;
        #pragma unroll
        for (int e = 0; e < NEXP; ++e) {
            float lv = lrow[e] + b[e];
            bool gt0 = lv > v0;
            bool gt1 = lv > v1;
            v1 = gt0 ? v0 : (gt1 ? lv : v1);
            i1 = gt0 ? i0 : (gt1 ? e  : i1);
            v0 = gt0 ? lv : v0;
            i0 = gt0 ? e  : i0;
        }
        // softmax over 2 (numerically safe: v0 >= v1)
        float e2 = __expf(v1 - v0);
        float g0 = 1.0f / (1.0f + e2);
        int tok = tokBase + tid;
        gates[tok * 2 + 0]  = g0;
        gates[tok * 2 + 1]  = e2 * g0;
        idxout[tok * 2 + 0] = i0;
        idxout[tok * 2 + 1] = i1;
    }
}

extern "C" void kernel_launch(void* const* d_in, const int* in_sizes, int n_in,
                              void* d_out, int out_size, void* d_ws, size_t ws_size,
                              hipStream_t stream)
{
    (void)in_sizes; (void)n_in; (void)d_ws; (void)ws_size; (void)out_size;
    const float* x = (const float*)d_in[0];   // [8192, 4096]
    const float* w = (const float*)d_in[1];   // [64, 4096]
    const float* b = (const float*)d_in[2];   // [64]

    // d_out: [gate_scores 8192*2 f32][topk_idx 8192*2 i32-bits], concatenated flat
    float* gatesp = (float*)d_out;
    int*   idxp   = (int*)d_out + TOKENS * 2;

    dim3 grid(TOKENS / BM), block(128);
    TopKGate_router_kernel<<<grid, block, 0, stream>>>(x, w, b, gatesp, idxp);
}